// WaveletDenoiseLayer_8065948582304
// MI455X (gfx1250) — compile-verified
//
#include <hip/hip_runtime.h>
#include <math.h>

typedef float v2f __attribute__((ext_vector_type(2)));
typedef float v8f __attribute__((ext_vector_type(8)));

#define NPLANES 96
#define HDIM 512
#define WDIM 512
#define NSUB 263                      // floor((512 + 16 - 1)/2)
#define DETN (NSUB * NSUB)            // 69169 per subband per plane
#define NBINS 4096
#define MEDIAN_TARGET 103754          // 1-based rank of median of 3*69169 = 207507
#define FWD_TAB 768                   // 12 chunks * 2(lg) * 16(lm) * 2(e)
#define INV_TAB 256                   // 4 chunks * 2 * 16 * 2

// db8 reconstruction filters; forward cross-correlation weights are exactly REC_LO/REC_HI
__constant__ float c_lo[16] = {
    0.05441584224308161f,  0.3128715909144659f,   0.6756307362980128f,
    0.5853546836548691f,  -0.015829105256023893f, -0.2840155429624281f,
    0.00047248457399797254f, 0.128747426620186f,  -0.01736930100202211f,
   -0.04408825393106472f,  0.013981027917015516f,  0.008746094047015655f,
   -0.00487035299301066f, -0.0003917403729959771f, 0.0006754494059985568f,
   -0.00011747678400228192f};
__constant__ float c_hi[16] = {
   -0.00011747678400228192f, -0.0006754494059985568f, -0.0003917403729959771f,
    0.00487035299301066f,     0.008746094047015655f,  -0.013981027917015516f,
   -0.04408825393106472f,     0.01736930100202211f,    0.128747426620186f,
   -0.00047248457399797254f, -0.2840155429624281f,     0.015829105256023893f,
    0.5853546836548691f,     -0.6756307362980128f,     0.3128715909144659f,
   -0.05441584224308161f};

__device__ __forceinline__ int reflect_idx(int i, int n) {
  i = (i < 0) ? (-1 - i) : i;
  i = (i >= n) ? (2 * n - 1 - i) : i;
  return i;
}

__device__ __forceinline__ v8f wmma4(v2f a, v2f b, v8f c) {
  // fp32 WMMA: D(16x16) += A(16x4) * B(4x16)
  return __builtin_amdgcn_wmma_f32_16x16x4_f32(false, a, false, b, (short)0, c,
                                               false, false);
}

__device__ __forceinline__ v2f ldfrag(const float* __restrict__ p, int idx) {
  return *(const v2f*)(p + idx);    // 8-byte aligned fragment load
}

// ------ K0: zero max/hist scratch + build filter fragment tables ------------
// Fragment layout: idx = ((chunk*2 + lg)*16 + lm)*2 + e  -> lane fragment is a
// contiguous float2, so hot kernels do one coalesced b64 load per fragment.
__global__ void k0_init(unsigned* __restrict__ maxabs,
                        unsigned* __restrict__ hist,
                        float* __restrict__ fwd_lo, float* __restrict__ fwd_hi,
                        float* __restrict__ inv_lo, float* __restrict__ inv_hi) {
  const int i = blockIdx.x * blockDim.x + threadIdx.x;
  if (i < NPLANES) maxabs[i] = 0u;
  if (i < NPLANES * NBINS) hist[i] = 0u;
  if (i < FWD_TAB) {
    const int e = i & 1, lm = (i >> 1) & 15, rest = i >> 5;  // rest = 2*kk+lg
    const int j = 2 * rest + e;        // j = 4*kk + 2*lg + e
    const int t = j - 2 * lm;          // forward band: B[j, n] = W[j - 2n]
    fwd_lo[i] = (t >= 0 && t < 16) ? c_lo[t] : 0.0f;
    fwd_hi[i] = (t >= 0 && t < 16) ? c_hi[t] : 0.0f;
  }
  if (i < INV_TAB) {
    const int e = i & 1, lm = (i >> 1) & 15, rest = i >> 5;
    const int kl = 2 * rest + e;
    const int tt = 2 * kl + 1 - lm;    // inverse band: F[i,k] = REC[15-(2k+1-i)]
    inv_lo[i] = (tt >= 0 && tt < 16) ? c_lo[15 - tt] : 0.0f;
    inv_hi[i] = (tt >= 0 && tt < 16) ? c_hi[15 - tt] : 0.0f;
  }
}

// ---------------- K1: forward DWT along W (rows), NHWC gather ----------------
__global__ void k1_row_dwt(const float* __restrict__ x,
                           const float* __restrict__ fwd_lo,
                           const float* __restrict__ fwd_hi,
                           float* __restrict__ lo, float* __restrict__ hi) {
  const int o0 = blockIdx.x * 16;   // output column tile (w')
  const int h0 = blockIdx.y * 16;   // row tile
  const int m  = blockIdx.z;
  const int b = m / 3, c = m % 3;
  const int lane = threadIdx.x, lm = lane & 15, lg = lane >> 4;
  const int fidx = (lg * 16 + lm) * 2;
  const int row = h0 + lm;
  const size_t rowbase = ((size_t)(b * HDIM + row)) * WDIM;
  v8f dlo = {}; v8f dhi = {};
  for (int kk = 0; kk < 12; ++kk) {         // K = 48 (window span 46, padded)
    const int j0 = 4 * kk + 2 * lg;
    v2f a;
#pragma unroll
    for (int e = 0; e < 2; ++e) {
      const int q = 2 * o0 + j0 + e;        // padded-signal index
      const int wc = reflect_idx(q - 14, WDIM);
      a[e] = x[(rowbase + wc) * 3 + c];     // A: data patch (M=row, K=j)
    }
    const v2f blo = ldfrag(fwd_lo, kk * 64 + fidx);
    const v2f bhi = ldfrag(fwd_hi, kk * 64 + fidx);
    dlo = wmma4(a, blo, dlo);
    dhi = wmma4(a, bhi, dhi);
  }
  const int wo = o0 + lm;
  if (wo < NSUB) {
#pragma unroll
    for (int v = 0; v < 8; ++v) {
      const int hh = h0 + v + 8 * lg;
      lo[((size_t)m * HDIM + hh) * NSUB + wo] = dlo[v];
      hi[((size_t)m * HDIM + hh) * NSUB + wo] = dhi[v];
    }
  }
}

// ------------- K2: forward DWT along H, output transposed [m,w',h''] --------
__global__ void k2_col_dwt(const float* __restrict__ lo,
                           const float* __restrict__ hi,
                           const float* __restrict__ fwd_lo,
                           const float* __restrict__ fwd_hi,
                           float* __restrict__ ll, float* __restrict__ lh,
                           float* __restrict__ hl, float* __restrict__ hh) {
  const int ho0 = blockIdx.x * 16;  // output index along transformed H
  const int w0  = blockIdx.y * 16;  // w' tile
  const int z = blockIdx.z;
  const int m = z % NPLANES;
  const bool isHi = z >= NPLANES;
  const float* src = isHi ? hi : lo;
  float* dstA = isHi ? hl : ll;
  float* dstB = isHi ? hh : lh;
  const int lane = threadIdx.x, lm = lane & 15, lg = lane >> 4;
  const int fidx = (lg * 16 + lm) * 2;
  // clamp: out-of-range w' only feeds D rows that are never stored
  const int wp = min(w0 + lm, NSUB - 1);
  v8f dA = {}; v8f dB = {};
  for (int kk = 0; kk < 12; ++kk) {
    const int j0 = 4 * kk + 2 * lg;
    v2f a;
#pragma unroll
    for (int e = 0; e < 2; ++e) {
      const int q = 2 * ho0 + j0 + e;
      const int hr = reflect_idx(q - 14, HDIM);
      a[e] = src[((size_t)m * HDIM + hr) * NSUB + wp];
    }
    const v2f blo = ldfrag(fwd_lo, kk * 64 + fidx);
    const v2f bhi = ldfrag(fwd_hi, kk * 64 + fidx);
    dA = wmma4(a, blo, dA);
    dB = wmma4(a, bhi, dB);
  }
  const int ho = ho0 + lm;
  if (ho < NSUB) {
#pragma unroll
    for (int v = 0; v < 8; ++v) {
      const int wpp = w0 + v + 8 * lg;
      if (wpp < NSUB) {
        dstA[((size_t)m * NSUB + wpp) * NSUB + ho] = dA[v];
        dstB[((size_t)m * NSUB + wpp) * NSUB + ho] = dB[v];
      }
    }
  }
}

// ---------------- K3a: per-plane max |detail| ----------------
__global__ void k3a_maxabs(const float* __restrict__ lh,
                           const float* __restrict__ hl,
                           const float* __restrict__ hh,
                           unsigned* __restrict__ maxabs) {
  __shared__ float red[256];
  const int m = blockIdx.y;
  const size_t base = (size_t)m * DETN;
  float mx = 0.0f;
  for (int i = blockIdx.x * blockDim.x + threadIdx.x; i < DETN;
       i += gridDim.x * blockDim.x) {
    mx = fmaxf(mx, fabsf(lh[base + i]));
    mx = fmaxf(mx, fabsf(hl[base + i]));
    mx = fmaxf(mx, fabsf(hh[base + i]));
  }
  red[threadIdx.x] = mx;
  __syncthreads();
  for (int s = 128; s > 0; s >>= 1) {
    if (threadIdx.x < s)
      red[threadIdx.x] = fmaxf(red[threadIdx.x], red[threadIdx.x + s]);
    __syncthreads();
  }
  if (threadIdx.x == 0) atomicMax(&maxabs[m], __float_as_uint(red[0]));
}

// ---------------- K3b: per-plane histogram of |detail| (LDS bins) -----------
__global__ void k3b_hist(const float* __restrict__ lh,
                         const float* __restrict__ hl,
                         const float* __restrict__ hh,
                         const unsigned* __restrict__ maxabs,
                         unsigned* __restrict__ hist) {
  __shared__ unsigned hs[NBINS];
  const int m = blockIdx.y;
  for (int j = threadIdx.x; j < NBINS; j += blockDim.x) hs[j] = 0u;
  __syncthreads();
  const float mv = fmaxf(__uint_as_float(maxabs[m]), 1e-30f);
  const float scale = (float)NBINS / mv;
  const size_t base = (size_t)m * DETN;
  for (int i = blockIdx.x * blockDim.x + threadIdx.x; i < DETN;
       i += gridDim.x * blockDim.x) {
    const int b0 = min(NBINS - 1, (int)(fabsf(lh[base + i]) * scale));
    const int b1 = min(NBINS - 1, (int)(fabsf(hl[base + i]) * scale));
    const int b2 = min(NBINS - 1, (int)(fabsf(hh[base + i]) * scale));
    atomicAdd(&hs[b0], 1u);
    atomicAdd(&hs[b1], 1u);
    atomicAdd(&hs[b2], 1u);
  }
  __syncthreads();
  for (int j = threadIdx.x; j < NBINS; j += blockDim.x) {
    const unsigned cnt = hs[j];
    if (cnt) atomicAdd(&hist[m * NBINS + j], cnt);
  }
}

// ---------------- K3c: rank-select median -> universal threshold ------------
__global__ void k3c_threshold(const unsigned* __restrict__ maxabs,
                              const unsigned* __restrict__ hist,
                              float* __restrict__ thr) {
  const int m = blockIdx.x * blockDim.x + threadIdx.x;
  if (m >= NPLANES) return;
  const float mv = fmaxf(__uint_as_float(maxabs[m]), 1e-30f);
  unsigned cum = 0;
  int bin = NBINS - 1;
  for (int j = 0; j < NBINS; ++j) {
    cum += hist[m * NBINS + j];
    if (cum >= MEDIAN_TARGET) { bin = j; break; }
  }
  const float med = (bin + 0.5f) * mv / (float)NBINS;
  const float sigma = med / 0.6745f;
  thr[m] = sigma * sqrtf(2.0f * logf((float)(HDIM * WDIM)));
}

__device__ __forceinline__ float softthr(float v, float t) {
  const float s = fabsf(v) - t;
  return (s > 0.0f) ? copysignf(s, v) : 0.0f;
}

// -------- K4: inverse DWT along H with fused soft-threshold -> rl/rh --------
__global__ void k4_icol(const float* __restrict__ ll,
                        const float* __restrict__ lh,
                        const float* __restrict__ hl,
                        const float* __restrict__ hh,
                        const float* __restrict__ inv_lo,
                        const float* __restrict__ inv_hi,
                        const float* __restrict__ thr,
                        float* __restrict__ rl, float* __restrict__ rh) {
  const int w0 = blockIdx.x * 16;   // w' tile
  const int i0 = blockIdx.y * 16;   // output-h tile (even)
  const int z = blockIdx.z;
  const int m = z % NPLANES;
  const bool isHi = z >= NPLANES;
  const float* sA = isHi ? hl : ll;   // approx path (hl is a detail: threshold)
  const float* sD = isHi ? hh : lh;   // detail path (always threshold)
  float* dst = isHi ? rh : rl;
  const int lane = threadIdx.x, lm = lane & 15, lg = lane >> 4;
  const int fidx = (lg * 16 + lm) * 2;
  const int k0 = i0 >> 1;
  const float t = thr[m];
  const int wp = w0 + lm;
  const bool wok = wp < NSUB;       // out-of-range w' -> D column never stored
  const size_t rowb = ((size_t)m * NSUB + min(wp, NSUB - 1)) * NSUB;
  v8f d = {};
  for (int kk = 0; kk < 4; ++kk) {          // K = 16 coefficients
    const v2f aLo = ldfrag(inv_lo, kk * 64 + fidx);  // A: inverse filter
    const v2f aHi = ldfrag(inv_hi, kk * 64 + fidx);
    v2f bA, bD;
#pragma unroll
    for (int e = 0; e < 2; ++e) {
      const int kl = 4 * kk + 2 * lg + e;
      const int k = k0 + kl;
      const int kc = min(k, NSUB - 1);
      float va = sA[rowb + kc];              // B: data (K=kl, N=w')
      float vd = sD[rowb + kc];
      if (isHi) va = softthr(va, t);
      vd = softthr(vd, t);
      bA[e] = (k < NSUB) ? va : 0.0f;        // zero K-tail, branch-free
      bD[e] = (k < NSUB) ? vd : 0.0f;
    }
    d = wmma4(aLo, bA, d);
    d = wmma4(aHi, bD, d);
  }
  if (wok) {
#pragma unroll
    for (int v = 0; v < 8; ++v) {
      const int hrow = i0 + v + 8 * lg;
      dst[((size_t)m * HDIM + hrow) * NSUB + wp] = d[v];
    }
  }
}

// -------- K5: inverse DWT along W, scatter to NHWC output -------------------
__global__ void k5_irow(const float* __restrict__ rl,
                        const float* __restrict__ rh,
                        const float* __restrict__ inv_lo,
                        const float* __restrict__ inv_hi,
                        float* __restrict__ out) {
  const int wo0 = blockIdx.x * 16;  // output-w tile (even)
  const int h0  = blockIdx.y * 16;  // row tile
  const int m = blockIdx.z;
  const int b = m / 3, c = m % 3;
  const int lane = threadIdx.x, lm = lane & 15, lg = lane >> 4;
  const int fidx = (lg * 16 + lm) * 2;
  const int k0 = wo0 >> 1;
  const int hrow = h0 + lm;
  const size_t rbase = ((size_t)m * HDIM + hrow) * NSUB;
  v8f d = {};
  for (int kk = 0; kk < 4; ++kk) {
    v2f aL, aH;
#pragma unroll
    for (int e = 0; e < 2; ++e) {
      const int kl = 4 * kk + 2 * lg + e;
      const int k = k0 + kl;
      const int kc = min(k, NSUB - 1);
      const float vl = rl[rbase + kc];       // A: data (M=h, K=kl)
      const float vh = rh[rbase + kc];
      aL[e] = (k < NSUB) ? vl : 0.0f;
      aH[e] = (k < NSUB) ? vh : 0.0f;
    }
    const v2f bLo = ldfrag(inv_lo, kk * 64 + fidx);  // B: inverse filter
    const v2f bHi = ldfrag(inv_hi, kk * 64 + fidx);
    d = wmma4(aL, bLo, d);
    d = wmma4(aH, bHi, d);
  }
#pragma unroll
  for (int v = 0; v < 8; ++v) {
    const int hh = h0 + v + 8 * lg;
    const int w = wo0 + lm;
    out[(((size_t)b * HDIM + hh) * WDIM + w) * 3 + c] = d[v];
  }
}

extern "C" void kernel_launch(void* const* d_in, const int* in_sizes, int n_in,
                              void* d_out, int out_size, void* d_ws,
                              size_t ws_size, hipStream_t stream) {
  (void)in_sizes; (void)n_in; (void)out_size; (void)ws_size;
  const float* x = (const float*)d_in[0];
  float* out = (float*)d_out;
  float* ws = (float*)d_ws;

  size_t off = 0;
  float* lo = ws + off; off += (size_t)NPLANES * HDIM * NSUB;
  float* hi = ws + off; off += (size_t)NPLANES * HDIM * NSUB;
  float* ll = ws + off; off += (size_t)NPLANES * DETN;
  float* lh = ws + off; off += (size_t)NPLANES * DETN;
  float* hl = ws + off; off += (size_t)NPLANES * DETN;
  float* hh = ws + off; off += (size_t)NPLANES * DETN;
  float* fwd_lo = ws + off; off += FWD_TAB;
  float* fwd_hi = ws + off; off += FWD_TAB;
  float* inv_lo = ws + off; off += INV_TAB;
  float* inv_hi = ws + off; off += INV_TAB;
  unsigned* maxabs = (unsigned*)(ws + off); off += NPLANES;
  unsigned* hist   = (unsigned*)(ws + off); off += (size_t)NPLANES * NBINS;
  float* thr = ws + off; off += NPLANES;
  float* rl = lo;   // reuse: lo/hi consumed by K2 before K4 writes rl/rh
  float* rh = hi;

  const dim3 wblk(32);
  k0_init<<<dim3((NPLANES * NBINS + 255) / 256), dim3(256), 0, stream>>>(
      maxabs, hist, fwd_lo, fwd_hi, inv_lo, inv_hi);
  k1_row_dwt<<<dim3(17, 32, NPLANES), wblk, 0, stream>>>(x, fwd_lo, fwd_hi, lo, hi);
  k2_col_dwt<<<dim3(17, 17, 2 * NPLANES), wblk, 0, stream>>>(
      lo, hi, fwd_lo, fwd_hi, ll, lh, hl, hh);
  k3a_maxabs<<<dim3(17, NPLANES), dim3(256), 0, stream>>>(lh, hl, hh, maxabs);
  k3b_hist<<<dim3(17, NPLANES), dim3(256), 0, stream>>>(lh, hl, hh, maxabs, hist);
  k3c_threshold<<<dim3(3), dim3(32), 0, stream>>>(maxabs, hist, thr);
  k4_icol<<<dim3(17, 32, 2 * NPLANES), wblk, 0, stream>>>(
      ll, lh, hl, hh, inv_lo, inv_hi, thr, rl, rh);
  k5_irow<<<dim3(32, 32, NPLANES), wblk, 0, stream>>>(rl, rh, inv_lo, inv_hi, out);
}